// CRF_14800457302425
// MI455X (gfx1250) — compile-verified
//
#include <hip/hip_runtime.h>
#include <hip/hip_bf16.h>
#include <math.h>

typedef float v2f __attribute__((ext_vector_type(2)));
typedef float v8f __attribute__((ext_vector_type(8)));

#define NUM_TAGS  128
#define START_TAG 1
#define END_TAG   2
#define NEGV      (-10000.0f)
#define BATCH     256
#define SEQ       512
#define TILE_M    16
#define ET_STRIDE 132   // padded row stride (floats) -> conflict-free LDS b64 reads
#define AE_STRIDE 132

// ---------------------------------------------------------------------------
// Kernel 1: per-column max of transitions + exp'd, transposed transitions.
//   ctm[j]      = max_i trans[i][j]
//   etT[j][i]   = exp(trans[i][j] - ctm[j])   (column-major for B-fragment loads)
// ---------------------------------------------------------------------------
__global__ void crf_prep(const float* __restrict__ trans,
                         float* __restrict__ etT,
                         float* __restrict__ ctm)
{
    const int j = threadIdx.x;                 // 0..127
    float m = -INFINITY;
    for (int i = 0; i < NUM_TAGS; ++i) m = fmaxf(m, trans[i * NUM_TAGS + j]);
    ctm[j] = m;
    for (int i = 0; i < NUM_TAGS; ++i)
        etT[j * NUM_TAGS + i] = __expf(trans[i * NUM_TAGS + j] - m);
}

// ---------------------------------------------------------------------------
// Kernel 2: forward algorithm. One workgroup = 16 batch rows, 8 waves.
// Per step: exp-space GEMM (256x128 @ 128x128 over the grid) on the fp32
// matrix pipe: each wave = one 16x16 tile, K=128 via 32x V_WMMA_F32_16X16X4_F32
// split over 4 independent accumulators to pipeline the WMMA unit.
// ---------------------------------------------------------------------------
__global__ __launch_bounds__(256, 1) void crf_forward(
    const float* __restrict__ emis,            // [B, S, T]
    const unsigned char* __restrict__ mask,    // [B, S] (bool)
    const float* __restrict__ trans,           // [T, T]
    const float* __restrict__ etT,             // [T, T] exp'd, transposed
    const float* __restrict__ ctm,             // [T]
    float* __restrict__ out)
{
    __shared__ float sEt[NUM_TAGS * ET_STRIDE];    // 67.6 KB: exp(trans)^T
    __shared__ float sAlpha[TILE_M * NUM_TAGS];    //  8   KB: alpha state
    __shared__ float sAexp[TILE_M * AE_STRIDE];    //  8.4 KB: exp(alpha - m)
    __shared__ float sCtm[NUM_TAGS];
    __shared__ float sMrow[TILE_M];
    __shared__ unsigned int sMask[TILE_M];

    const int tid  = threadIdx.x;
    const int wave = tid >> 5;                 // 0..7  -> column tile
    const int lane = tid & 31;
    const int hi   = lane >> 4;                // 0/1: K-half within fragment
    const int ln   = lane & 15;                // M row (A) / N col (B,C)
    const int b0   = blockIdx.x * TILE_M;
    const int jg   = wave * 16 + ln;           // global tag column of this lane

    // Stage exp'd transitions into LDS (padded stride), load ctm, init alpha.
    for (int idx = tid; idx < NUM_TAGS * NUM_TAGS; idx += 256) {
        const int j = idx >> 7, i = idx & 127;
        sEt[j * ET_STRIDE + i] = etT[idx];
    }
    if (tid < NUM_TAGS) sCtm[tid] = ctm[tid];
    for (int idx = tid; idx < TILE_M * NUM_TAGS; idx += 256) {
        const int j = idx & 127;
        sAlpha[idx] = (j == START_TAG) ? 0.0f : NEGV;
    }
    __syncthreads();

    const int r16 = tid >> 4;                  // phase-1 row (0..15)
    const int c16 = (tid & 15) * 8;            // phase-1 column group

    for (int t = 0; t < SEQ; ++t) {
        // ---- phase 1: per-row max, exponentiate alpha into WMMA-A staging --
        float av[8];
        float m = -INFINITY;
        #pragma unroll
        for (int q = 0; q < 8; ++q) {
            av[q] = sAlpha[r16 * NUM_TAGS + c16 + q];
            m = fmaxf(m, av[q]);
        }
        #pragma unroll
        for (int off = 8; off >= 1; off >>= 1)
            m = fmaxf(m, __shfl_xor(m, off, 32));
        #pragma unroll
        for (int q = 0; q < 8; ++q)
            sAexp[r16 * AE_STRIDE + c16 + q] = __expf(av[q] - m);
        if ((tid & 15) == 0) sMrow[r16] = m;
        if (tid < TILE_M) sMask[tid] = mask[(size_t)(b0 + tid) * SEQ + t];
        // Prefetch next step's emission tile (uniform guard, near scope).
        if (t + 1 < SEQ)
            __builtin_prefetch(
                &emis[((size_t)(b0 + r16) * SEQ + (t + 1)) * NUM_TAGS + c16],
                0, 3);
        __syncthreads();

        // ---- issue this step's emission loads early: hide HBM latency under
        //      the WMMA chain (consumed in phase 3).
        float ev[8];
        #pragma unroll
        for (int v = 0; v < 8; ++v) {
            const int r = v + 8 * hi;                  // C layout: M = v + 8*hi
            ev[v] = emis[((size_t)(b0 + r) * SEQ + t) * NUM_TAGS + jg];
        }

        // ---- phase 2: 16x16 fp32 WMMA tile, K = 128 in 32 steps of 4,
        //      4 independent accumulators to break the C->C RAW chain. -------
        v8f c0 = {}, c1 = {}, c2 = {}, c3 = {};
        #pragma unroll 2
        for (int k = 0; k < 32; k += 4) {
            const int kk = 4 * k + 2 * hi;
            // A 16x4 f32 fragments: lanes 0-15 K={0,1}, lanes 16-31 K={2,3}
            const v2f a0 = *(const v2f*)&sAexp[ln * AE_STRIDE + kk];
            const v2f a1 = *(const v2f*)&sAexp[ln * AE_STRIDE + kk + 4];
            const v2f a2 = *(const v2f*)&sAexp[ln * AE_STRIDE + kk + 8];
            const v2f a3 = *(const v2f*)&sAexp[ln * AE_STRIDE + kk + 12];
            // B 4x16 f32 fragments: same K striping, N = lane&15
            const v2f b0 = *(const v2f*)&sEt[jg * ET_STRIDE + kk];
            const v2f b1 = *(const v2f*)&sEt[jg * ET_STRIDE + kk + 4];
            const v2f b2 = *(const v2f*)&sEt[jg * ET_STRIDE + kk + 8];
            const v2f b3 = *(const v2f*)&sEt[jg * ET_STRIDE + kk + 12];
            c0 = __builtin_amdgcn_wmma_f32_16x16x4_f32(
                     false, a0, false, b0, (short)0, c0, false, false);
            c1 = __builtin_amdgcn_wmma_f32_16x16x4_f32(
                     false, a1, false, b1, (short)0, c1, false, false);
            c2 = __builtin_amdgcn_wmma_f32_16x16x4_f32(
                     false, a2, false, b2, (short)0, c2, false, false);
            c3 = __builtin_amdgcn_wmma_f32_16x16x4_f32(
                     false, a3, false, b3, (short)0, c3, false, false);
        }
        const v8f c = (c0 + c1) + (c2 + c3);

        // ---- phase 3: back to log-space, add emissions, update alpha -------
        const float cj = sCtm[jg];
        #pragma unroll
        for (int v = 0; v < 8; ++v) {
            const int r = v + 8 * hi;
            const float nxt = sMrow[r] + cj + ev[v] + __logf(c[v]);
            const float old = sAlpha[r * NUM_TAGS + jg];
            sAlpha[r * NUM_TAGS + jg] = sMask[r] ? nxt : old;
        }
        __syncthreads();
    }

    // ---- final: fwd_b = logsumexp_j(alpha[b][j] + trans[j][END]) -----------
    float term[8];
    float m2 = -INFINITY;
    #pragma unroll
    for (int q = 0; q < 8; ++q) {
        const int j = c16 + q;
        term[q] = sAlpha[r16 * NUM_TAGS + j] + trans[j * NUM_TAGS + END_TAG];
        m2 = fmaxf(m2, term[q]);
    }
    #pragma unroll
    for (int off = 8; off >= 1; off >>= 1)
        m2 = fmaxf(m2, __shfl_xor(m2, off, 32));
    float s = 0.0f;
    #pragma unroll
    for (int q = 0; q < 8; ++q) s += __expf(term[q] - m2);
    #pragma unroll
    for (int off = 8; off >= 1; off >>= 1) s += __shfl_xor(s, off, 32);
    if ((tid & 15) == 0) atomicAdd(out, m2 + logf(s));
}

// ---------------------------------------------------------------------------
// Kernel 3: gold path score (subtracted). One block per batch row.
// ---------------------------------------------------------------------------
__global__ void crf_gold(const float* __restrict__ emis,
                         const unsigned char* __restrict__ mask,
                         const int* __restrict__ tags,
                         const float* __restrict__ trans,
                         float* __restrict__ out)
{
    const int b = blockIdx.x;
    const int tid = threadIdx.x;               // 128 threads
    __shared__ float part[4];

    float local = 0.0f;
    for (int t = tid; t < SEQ; t += 128) {
        const int tg = tags[b * SEQ + t];
        local += emis[((size_t)b * SEQ + t) * NUM_TAGS + tg];
        if (t >= 1 && mask[(size_t)b * SEQ + t])
            local += trans[tags[b * SEQ + t - 1] * NUM_TAGS + tg];
    }
    if (tid == 0) {
        local += trans[START_TAG * NUM_TAGS + tags[b * SEQ]];
        if (mask[(size_t)b * SEQ + SEQ - 1])
            local += trans[tags[b * SEQ + SEQ - 1] * NUM_TAGS + END_TAG];
    }
    #pragma unroll
    for (int off = 16; off >= 1; off >>= 1) local += __shfl_xor(local, off, 32);
    if ((tid & 31) == 0) part[tid >> 5] = local;
    __syncthreads();
    if (tid == 0) atomicAdd(out, -(part[0] + part[1] + part[2] + part[3]));
}

// ---------------------------------------------------------------------------
extern "C" void kernel_launch(void* const* d_in, const int* in_sizes, int n_in,
                              void* d_out, int out_size, void* d_ws, size_t ws_size,
                              hipStream_t stream)
{
    (void)in_sizes; (void)n_in; (void)ws_size;
    const float*         emis  = (const float*)d_in[0];
    const unsigned char* mask  = (const unsigned char*)d_in[1];
    const int*           tags  = (const int*)d_in[2];
    const float*         trans = (const float*)d_in[3];
    float* out = (float*)d_out;

    float* etT = (float*)d_ws;                        // 128*128 floats (64 KB)
    float* ctm = etT + NUM_TAGS * NUM_TAGS;           // 128 floats

    hipMemsetAsync(d_out, 0, sizeof(float) * out_size, stream);
    crf_prep   <<<1,               NUM_TAGS, 0, stream>>>(trans, etT, ctm);
    crf_forward<<<BATCH / TILE_M,  256,      0, stream>>>(emis, mask, trans, etT, ctm, out);
    crf_gold   <<<BATCH,           128,      0, stream>>>(emis, mask, tags, trans, out);
}